// DiceLoss_8839042695183
// MI455X (gfx1250) — compile-verified
//
#include <hip/hip_runtime.h>
#include <hip/hip_bf16.h>

typedef float v2f __attribute__((ext_vector_type(2)));
typedef float v8f __attribute__((ext_vector_type(8)));

#define NCLS     11
#define HW       (512 * 512)        // 262144 pixels per image
#define NB       8
#define NGRP     ((NB * HW) / 4)    // 524288 float4 groups
#define GRPS_IMG (HW / 4)           // 65536 = 2^16
#define P1BLOCKS 1024
#define P1TPB    256
#define CPAD     32                 // padded partial slots per block

// ---------------------------------------------------------------------------
// Pass 1: streaming softmax + per-class accumulation, per-block partials.
// partials[block][0..10]  = sum of probs[c] over pixels with target==c (intersection)
// partials[block][11..21] = sum of probs[c] + count(target==c)        (cardinality)
// partials[block][22..31] = 0 (padding for the WMMA reduction)
// ---------------------------------------------------------------------------
__global__ __launch_bounds__(P1TPB) void dice_partials(
    const float* __restrict__ logits,
    const int*   __restrict__ targets,
    float*       __restrict__ partials) {

  float accI[NCLS];
  float accK[NCLS];
#pragma unroll
  for (int c = 0; c < NCLS; ++c) { accI[c] = 0.0f; accK[c] = 0.0f; }

  const int tid      = blockIdx.x * blockDim.x + threadIdx.x;
  const int nthreads = gridDim.x * blockDim.x;

  for (int g = tid; g < NGRP; g += nthreads) {
    const int b   = g >> 16;                 // g / GRPS_IMG
    const int hw4 = (g & (GRPS_IMG - 1)) << 2;

    const float* base = logits + (size_t)b * NCLS * HW + hw4;

    float4 x[NCLS];
#pragma unroll
    for (int c = 0; c < NCLS; ++c)
      x[c] = *reinterpret_cast<const float4*>(base + (size_t)c * HW);

    const int4 t4 = *reinterpret_cast<const int4*>(targets + (size_t)b * HW + hw4);
    const int tg[4] = {t4.x, t4.y, t4.z, t4.w};

#pragma unroll
    for (int j = 0; j < 4; ++j) {
      float v[NCLS];
#pragma unroll
      for (int c = 0; c < NCLS; ++c)
        v[c] = (j == 0) ? x[c].x : (j == 1) ? x[c].y : (j == 2) ? x[c].z : x[c].w;

      float m = v[0];
#pragma unroll
      for (int c = 1; c < NCLS; ++c) m = fmaxf(m, v[c]);

      float s = 0.0f;
#pragma unroll
      for (int c = 0; c < NCLS; ++c) { v[c] = __expf(v[c] - m); s += v[c]; }

      const float inv    = __builtin_amdgcn_rcpf(s);
      const int   t      = tg[j];
      const float mvalid = (t != 255) ? 1.0f : 0.0f;   // ignore-index mask
      const float scale  = inv * mvalid;

#pragma unroll
      for (int c = 0; c < NCLS; ++c) {
        const float p   = v[c] * scale;                // masked prob
        const float ind = (c == t) ? 1.0f : 0.0f;      // never true when t==255
        accK[c] += p + ind;                            // cardinality contribution
        accI[c] += p * ind;                            // intersection contribution
      }
    }
  }

  // ---- wave32 reduction (warpSize == 32 on gfx1250) ----
#pragma unroll
  for (int c = 0; c < NCLS; ++c) {
#pragma unroll
    for (int off = 16; off > 0; off >>= 1) {
      accI[c] += __shfl_xor(accI[c], off, 32);
      accK[c] += __shfl_xor(accK[c], off, 32);
    }
  }

  __shared__ float lds[P1TPB / 32][2 * NCLS];
  const int lane = threadIdx.x & 31;
  const int wave = threadIdx.x >> 5;
  if (lane == 0) {
#pragma unroll
    for (int c = 0; c < NCLS; ++c) {
      lds[wave][c]        = accI[c];
      lds[wave][NCLS + c] = accK[c];
    }
  }
  __syncthreads();

  if (threadIdx.x < CPAD) {
    float s = 0.0f;
    if (threadIdx.x < 2 * NCLS) {
#pragma unroll
      for (int w = 0; w < P1TPB / 32; ++w) s += lds[w][threadIdx.x];
    }
    partials[(size_t)blockIdx.x * CPAD + threadIdx.x] = s;  // slots 22..31 get 0
  }
}

// ---------------------------------------------------------------------------
// Pass 2: reduce partials[1024][32] over blocks with V_WMMA_F32_16X16X4_F32.
// B = ones  =>  D[m][n] = sum_k A[m][k] + C[m][n]  (4 chunks per issue).
// A layout (ISA 7.12.2): lane m (0..15) holds A[m][0..1] in VGPR0..1,
//                        lane m+16     holds A[m][2..3].
// Two waves: wave w reduces padded classes w*16 .. w*16+15.
// ---------------------------------------------------------------------------
__global__ __launch_bounds__(64) void dice_finalize(
    const float* __restrict__ partials,
    float*       __restrict__ out) {

  __shared__ float sums[CPAD];

  const int lane = threadIdx.x & 31;
  const int wave = threadIdx.x >> 5;          // 0 or 1
  const int cls  = (lane & 15) + wave * 16;   // padded class owned by this lane
  const int coff = (lane < 16) ? 0 : 2;       // lanes 16..31 supply chunks i+2, i+3

  v8f acc = {0.f, 0.f, 0.f, 0.f, 0.f, 0.f, 0.f, 0.f};
  const v2f bones = {1.0f, 1.0f};

  for (int i = 0; i < P1BLOCKS; i += 4) {
    v2f a;
    a.x = partials[(size_t)(i + coff + 0) * CPAD + cls];
    a.y = partials[(size_t)(i + coff + 1) * CPAD + cls];
    // D = A(16x4) * ones(4x16) + C : accumulates 4 block-partials per class
    acc = __builtin_amdgcn_wmma_f32_16x16x4_f32(
        /*neg_a=*/false, a, /*neg_b=*/false, bones,
        /*c_mod=*/(short)0, acc, /*reuse_a=*/false, /*reuse_b=*/false);
  }

  // D layout: VGPR r -> rows M=r (lanes 0..15) and M=r+8 (lanes 16..31); all
  // columns identical (B was ones), so read column n=0 from lane 0 / lane 16.
  if (lane == 0) {
#pragma unroll
    for (int m = 0; m < 8; ++m) sums[wave * 16 + m] = acc[m];
  }
  if (lane == 16) {
#pragma unroll
    for (int m = 0; m < 8; ++m) sums[wave * 16 + 8 + m] = acc[m];
  }
  __syncthreads();

  if (threadIdx.x == 0) {
    float dice = 0.0f;
#pragma unroll
    for (int c = 0; c < NCLS; ++c) {
      const float I = sums[c];
      const float K = sums[NCLS + c];
      dice += (2.0f * I + 1.0f) / (K + 1.0f);   // SMOOTH = 1.0
    }
    out[0] = 1.0f - dice / (float)NCLS;
  }
}

// ---------------------------------------------------------------------------
extern "C" void kernel_launch(void* const* d_in, const int* in_sizes, int n_in,
                              void* d_out, int out_size, void* d_ws, size_t ws_size,
                              hipStream_t stream) {
  const float* logits  = (const float*)d_in[0];
  const int*   targets = (const int*)d_in[1];
  float*       partials = (float*)d_ws;       // 1024 * 32 * 4 B = 128 KB

  dice_partials<<<P1BLOCKS, P1TPB, 0, stream>>>(logits, targets, partials);
  dice_finalize<<<1, 64, 0, stream>>>(partials, (float*)d_out);
}